// FractionalDerivative_62088047231681
// MI455X (gfx1250) — compile-verified
//
#include <hip/hip_runtime.h>

// FractionalDerivative (GL depthwise causal conv, K=10) for gfx1250 (MI455X).
// out[b,c,t] = sum_{k=0..9} w[c,k] * x[b,c,t-k], zero left-pad.
// Mapped to V_WMMA_F32_16X16X4_F32:
//   D[m][n] = out[t0+16m+n] = sum_{j'} A[m][j'] * B[j'][n]
//   A[m][j'] = x[t0+16m+15-j']   (j' = 0..27, 7 K=4 slices)
//   B[j'][n] = w[n+j'-15] if 0 <= n+j'-15 < 10 else 0
// fp32 WMMA keeps reference precision; problem is HBM-bandwidth bound
// (~537 MB moved vs 1.37 GFLOP -> 2.6 FLOP/B). Data path: b128 global loads
// -> LDS (single HBM touch per element) -> b64 LDS reads -> WMMA -> stores.

typedef __attribute__((ext_vector_type(2))) float v2f;
typedef __attribute__((ext_vector_type(8))) float v8f;

#define K_ORDER   10
#define TILE_T    2048          // outputs per 256-thread block (8 waves x 256)
#define TPB       256
#define LDS_PAD   16            // front halo pad (covers j' up to 27 at m=0)
#define LDS_ELEMS (TILE_T + LDS_PAD)        // 2064 floats
#define LDS_VEC4  (LDS_ELEMS / 4)           // 516 float4

__global__ __launch_bounds__(TPB)
void frac_deriv_wmma_f32(const float* __restrict__ x,
                         const float* __restrict__ alpha,
                         float* __restrict__ out,
                         int C, int T)
{
    __shared__ float4 tile4[LDS_VEC4];
    float* tile = (float*)tile4;

    const int tblk = blockIdx.x;          // T-tile index
    const int c    = blockIdx.y;          // channel
    const int b    = blockIdx.z;          // batch
    const long long base = ((long long)b * C + c) * (long long)T;
    const int tbase = tblk * TILE_T;

    // ---- Stage x[tbase-16 .. tbase+2047] into LDS with b128 loads. ----
    // tbase-16 is 16B aligned; any float4 containing a t<0 element is
    // entirely t<0 (t4 = tbase-16+4*i4), so the pad select is per-vector.
    const float4* x4 = (const float4*)(x + base);
    for (int i4 = threadIdx.x; i4 < LDS_VEC4; i4 += TPB) {
        const int t4 = tbase - LDS_PAD + 4 * i4;     // multiple of 4; < T
        float4 v = make_float4(0.0f, 0.0f, 0.0f, 0.0f);
        if (t4 >= 0) v = x4[t4 >> 2];
        tile4[i4] = v;
    }
    __syncthreads();

    // ---- GL weights w[0..9] from alpha[c] (exact reference recurrence). ----
    float aa = alpha[c];
    aa = fminf(fmaxf(aa, 0.01f), 1.99f);
    float w[K_ORDER];
    w[0] = 1.0f;
#pragma unroll
    for (int k = 1; k < K_ORDER; ++k)
        w[k] = w[k - 1] * (((float)(k - 1) - aa) / (float)k);

    const int lane = threadIdx.x & 31;
    const int wv   = threadIdx.x >> 5;    // wave id: owns 256-output subtile
    const int n    = lane & 15;           // B/D column, also A row m
    const int kh   = lane >> 4;           // K-half select (ISA A/B layout)

    // ---- Build banded B slices in registers (branch-free -> EXEC all 1s).
    // B layout (4x16 f32): VGPR e holds row K = 2*kh + e for this lane half.
    v2f bs[7];
#pragma unroll
    for (int s = 0; s < 7; ++s) {
#pragma unroll
        for (int e = 0; e < 2; ++e) {
            const int jp = 4 * s + 2 * kh + e;       // j' in 0..27
            const int kk = n + jp - 15;              // conv tap index
            bs[s][e] = (kk >= 0 && kk < K_ORDER) ? w[kk] : 0.0f;
        }
    }

    // ---- 7 x V_WMMA_F32_16X16X4_F32 accumulation. ----
    // A layout (16x4 f32): lane m (0..15 in each half) VGPR e = K 2*kh+e.
    // LDS index p = 256*wv + 16*m + 31 - j' (m == n here). The pair
    // (p, p-1) starts at even index 16n+30-2kh-4s -> one aligned b64 read.
    v8f acc = {};
    const float2* tile2 = (const float2*)tile;
    const int pair0 = (256 * wv + 16 * n + 30 - 2 * kh) >> 1;  // float2 index
#pragma unroll
    for (int s = 0; s < 7; ++s) {
        const float2 p = tile2[pair0 - 2 * s];       // {tile[lo], tile[lo+1]}
        v2f a;
        a[0] = p.y;                                  // j' = 4s + 2kh
        a[1] = p.x;                                  // j' = 4s + 2kh + 1
        acc = __builtin_amdgcn_wmma_f32_16x16x4_f32(
            /*neg_a=*/false, a, /*neg_b=*/false, bs[s],
            /*c_mod=*/(short)0, acc, /*reuse_a=*/false, /*reuse_b=*/false);
    }

    // ---- Store D per 16x16 f32 C/D layout: VGPR r -> row M = r + 8*kh. ----
    // Each store: lanes 0-15 and 16-31 each cover a contiguous 64B run.
    const long long obase = base + (long long)(tbase + 256 * wv);
#pragma unroll
    for (int r = 0; r < 8; ++r)
        out[obase + 16 * (r + 8 * kh) + n] = acc[r];
}

extern "C" void kernel_launch(void* const* d_in, const int* in_sizes, int n_in,
                              void* d_out, int out_size, void* d_ws, size_t ws_size,
                              hipStream_t stream)
{
    (void)n_in; (void)out_size; (void)d_ws; (void)ws_size;
    const float* x     = (const float*)d_in[0];
    const float* alpha = (const float*)d_in[1];
    float*       out   = (float*)d_out;

    const int C = in_sizes[1];                 // 512
    const int T = 8192;                        // per reference
    const int B = in_sizes[0] / (C * T);       // 16

    dim3 grid(T / TILE_T, C, B);               // (4, 512, 16)
    frac_deriv_wmma_f32<<<grid, TPB, 0, stream>>>(x, alpha, out, C, T);
}